// GNN_5153960755249
// MI455X (gfx1250) — compile-verified
//
#include <hip/hip_runtime.h>

#define N_NODES  100000
#define N_EDGES  1600000
#define N_GRAPHS 64

typedef __attribute__((ext_vector_type(2))) float v2f;
typedef __attribute__((ext_vector_type(8))) float v8f;

// ---------------------------------------------------------------------------
// degree / inverse-degree
// ---------------------------------------------------------------------------
__global__ void degree_kernel(const int* __restrict__ dst, float* __restrict__ deg, int nEdges) {
    int i = blockIdx.x * blockDim.x + threadIdx.x;
    if (i < nEdges) atomicAdd(&deg[dst[i]], 1.0f);
}

__global__ void deginv_kernel(float* __restrict__ deg, int n) {
    int i = blockIdx.x * blockDim.x + threadIdx.x;
    if (i < n) deg[i] = 1.0f / fmaxf(deg[i], 1.0f);
}

// ---------------------------------------------------------------------------
// edge scatter: msg[dst] += x[src]   (coalesced float4 per lane-quad)
// ---------------------------------------------------------------------------
template<int F>
__global__ void scatter_kernel(const float* __restrict__ x, const int* __restrict__ src,
                               const int* __restrict__ dst, float* __restrict__ msg, int nEdges) {
    const int QP = F / 4;
    long long idx = (long long)blockIdx.x * blockDim.x + threadIdx.x;
    int e = (int)(idx / QP);
    int q = (int)(idx % QP);
    if (e >= nEdges) return;
    int s = src[e], d = dst[e];
    const float4 v = ((const float4*)(x + (size_t)s * F))[q];
    float* o = msg + (size_t)d * F + (size_t)q * 4;
    atomicAdd(o + 0, v.x);
    atomicAdd(o + 1, v.y);
    atomicAdd(o + 2, v.z);
    atomicAdd(o + 3, v.w);
}

// ---------------------------------------------------------------------------
// Fused SAGE linear:
//   out = maybe_relu( (agg .* deginv) @ Wl^T + bias + x @ Wr^T )
// One wave per 16-node x 16-outchannel tile, f32 WMMA 16x16x4, K-loop.
//
// A-fragment (16x4 f32): lane L -> row m = L&15; VGPR pair holds
//   K = k0 + 2*(L>>4) + {0,1}  (ISA 7.12.2 32-bit A layout)
// B-fragment (4x16 f32): lane L -> col n = L&15; same K striping.
// C/D (16x16 f32, 8 VGPRs): VGPR r -> row r + 8*(L>>4), col L&15.
// ---------------------------------------------------------------------------
template<int K, int NOUT, bool RELU>
__global__ __launch_bounds__(256) void sage_wmma_kernel(
    const float* __restrict__ agg, const float* __restrict__ x,
    const float* __restrict__ Wl, const float* __restrict__ Wr,
    const float* __restrict__ bias, const float* __restrict__ deginv,
    float* __restrict__ out)
{
    const int lane = threadIdx.x & 31;
    const int wave = threadIdx.x >> 5;
    const int numOutTiles  = NOUT / 16;
    const int numNodeTiles = N_NODES / 16;   // 6250, exact
    int tile = blockIdx.x * (blockDim.x >> 5) + wave;
    if (tile >= numNodeTiles * numOutTiles) return;   // wave-uniform: EXEC stays all-1s
    const int nodeTile = tile / numOutTiles;
    const int outTile  = tile % numOutTiles;

    const int m   = lane & 15;
    const int hlf = lane >> 4;
    const int node = nodeTile * 16 + m;           // A row
    const int n    = outTile * 16 + (lane & 15);  // B col

    const float dinv = deginv[node];

    v8f c = {};
    // (agg * deg_inv) @ Wl^T
    for (int k0 = 0; k0 < K; k0 += 4) {
        const int kb = k0 + 2 * hlf;
        v2f a, bb;
        a.x  = agg[(size_t)node * K + kb]     * dinv;
        a.y  = agg[(size_t)node * K + kb + 1] * dinv;
        bb.x = Wl[(size_t)n * K + kb];
        bb.y = Wl[(size_t)n * K + kb + 1];
        c = __builtin_amdgcn_wmma_f32_16x16x4_f32(false, a, false, bb, (short)0, c, false, false);
    }
    // x @ Wr^T  (accumulate into same C)
    for (int k0 = 0; k0 < K; k0 += 4) {
        const int kb = k0 + 2 * hlf;
        v2f a, bb;
        a.x  = x[(size_t)node * K + kb];
        a.y  = x[(size_t)node * K + kb + 1];
        bb.x = Wr[(size_t)n * K + kb];
        bb.y = Wr[(size_t)n * K + kb + 1];
        c = __builtin_amdgcn_wmma_f32_16x16x4_f32(false, a, false, bb, (short)0, c, false, false);
    }

    const float bv = bias[n];
#pragma unroll
    for (int r = 0; r < 8; ++r) {
        int mm = r + 8 * hlf;
        float v = c[r] + bv;
        if (RELU) v = fmaxf(v, 0.0f);
        out[(size_t)(nodeTile * 16 + mm) * NOUT + n] = v;
    }
}

// ---------------------------------------------------------------------------
// global mean pool (atomic accumulate)
// ---------------------------------------------------------------------------
__global__ void pool_kernel(const float* __restrict__ h, const int* __restrict__ batch,
                            float* __restrict__ gsum, float* __restrict__ gcnt) {
    long long idx = (long long)blockIdx.x * blockDim.x + threadIdx.x;
    if (idx >= (long long)N_NODES * 64) return;
    int node = (int)(idx >> 6);
    int f    = (int)(idx & 63);
    int g    = batch[node];
    atomicAdd(&gsum[g * 64 + f], h[idx]);
    if (f == 0) atomicAdd(&gcnt[g], 1.0f);
}

// ---------------------------------------------------------------------------
// tiny MLP head: emb(64x64) -> 32 -> 32 -> 10, single block with LDS
// ---------------------------------------------------------------------------
__global__ __launch_bounds__(256) void head_kernel(
    const float* __restrict__ gsum, const float* __restrict__ gcnt,
    const float* __restrict__ Wl1, const float* __restrict__ bl1,
    const float* __restrict__ Wl2, const float* __restrict__ bl2,
    const float* __restrict__ Wl3, const float* __restrict__ bl3,
    float* __restrict__ out)
{
    __shared__ float emb[64 * 64];
    __shared__ float o1[64 * 32];
    __shared__ float o2[64 * 32];
    int tid = threadIdx.x;
    for (int i = tid; i < 64 * 64; i += 256) {
        int g = i >> 6;
        emb[i] = gsum[i] / fmaxf(gcnt[g], 1.0f);
    }
    __syncthreads();
    for (int i = tid; i < 64 * 32; i += 256) {
        int g = i >> 5, j = i & 31;
        float s = bl1[j];
        for (int k = 0; k < 64; ++k) s += emb[g * 64 + k] * Wl1[j * 64 + k];
        o1[i] = fmaxf(s, 0.0f);
    }
    __syncthreads();
    for (int i = tid; i < 64 * 32; i += 256) {
        int g = i >> 5, j = i & 31;
        float s = bl2[j];
        for (int k = 0; k < 32; ++k) s += o1[g * 32 + k] * Wl2[j * 32 + k];
        o2[i] = fmaxf(s, 0.0f);
    }
    __syncthreads();
    for (int i = tid; i < 64 * 10; i += 256) {
        int g = i / 10, j = i % 10;
        float s = bl3[j];
        for (int k = 0; k < 32; ++k) s += o2[g * 32 + k] * Wl3[j * 32 + k];
        out[i] = s;
    }
}

// ---------------------------------------------------------------------------
extern "C" void kernel_launch(void* const* d_in, const int* in_sizes, int n_in,
                              void* d_out, int out_size, void* d_ws, size_t ws_size,
                              hipStream_t stream) {
    const float* x    = (const float*)d_in[0];
    const int*   ei   = (const int*)d_in[1];
    const int*   batch= (const int*)d_in[2];
    const float* W1l  = (const float*)d_in[3];
    const float* b1   = (const float*)d_in[4];
    const float* W1r  = (const float*)d_in[5];
    const float* W2l  = (const float*)d_in[6];
    const float* b2   = (const float*)d_in[7];
    const float* W2r  = (const float*)d_in[8];
    const float* W3l  = (const float*)d_in[9];
    const float* b3   = (const float*)d_in[10];
    const float* W3r  = (const float*)d_in[11];
    const float* Wl1  = (const float*)d_in[12];
    const float* bl1  = (const float*)d_in[13];
    const float* Wl2  = (const float*)d_in[14];
    const float* bl2  = (const float*)d_in[15];
    const float* Wl3  = (const float*)d_in[16];
    const float* bl3  = (const float*)d_in[17];
    const int* src = ei;
    const int* dst = ei + N_EDGES;

    float* ws   = (float*)d_ws;
    float* deg  = ws;  ws += N_NODES;
    float* msg  = ws;  ws += (size_t)N_NODES * 64;
    float* h1   = ws;  ws += (size_t)N_NODES * 32;
    float* h2   = ws;  ws += (size_t)N_NODES * 48;
    float* h3   = ws;  ws += (size_t)N_NODES * 64;
    float* gsum = ws;  ws += 64 * 64;
    float* gcnt = ws;  ws += 64;

    const int B = 256;

    // degrees -> inverse degrees (in place)
    hipMemsetAsync(deg, 0, N_NODES * sizeof(float), stream);
    degree_kernel<<<(N_EDGES + B - 1) / B, B, 0, stream>>>(dst, deg, N_EDGES);
    deginv_kernel<<<(N_NODES + B - 1) / B, B, 0, stream>>>(deg, N_NODES);

    // layer 1: 64 -> 32
    hipMemsetAsync(msg, 0, (size_t)N_NODES * 64 * sizeof(float), stream);
    {
        long long tot = (long long)N_EDGES * (64 / 4);
        scatter_kernel<64><<<(int)((tot + B - 1) / B), B, 0, stream>>>(x, src, dst, msg, N_EDGES);
        int tiles = (N_NODES / 16) * (32 / 16);
        sage_wmma_kernel<64, 32, true><<<(tiles + 7) / 8, B, 0, stream>>>(msg, x, W1l, W1r, b1, deg, h1);
    }

    // layer 2: 32 -> 48
    hipMemsetAsync(msg, 0, (size_t)N_NODES * 32 * sizeof(float), stream);
    {
        long long tot = (long long)N_EDGES * (32 / 4);
        scatter_kernel<32><<<(int)((tot + B - 1) / B), B, 0, stream>>>(h1, src, dst, msg, N_EDGES);
        int tiles = (N_NODES / 16) * (48 / 16);
        sage_wmma_kernel<32, 48, true><<<(tiles + 7) / 8, B, 0, stream>>>(msg, h1, W2l, W2r, b2, deg, h2);
    }

    // layer 3: 48 -> 64 (no relu)
    hipMemsetAsync(msg, 0, (size_t)N_NODES * 48 * sizeof(float), stream);
    {
        long long tot = (long long)N_EDGES * (48 / 4);
        scatter_kernel<48><<<(int)((tot + B - 1) / B), B, 0, stream>>>(h2, src, dst, msg, N_EDGES);
        int tiles = (N_NODES / 16) * (64 / 16);
        sage_wmma_kernel<48, 64, false><<<(tiles + 7) / 8, B, 0, stream>>>(msg, h2, W3l, W3r, b3, deg, h3);
    }

    // pool + head
    hipMemsetAsync(gsum, 0, (64 * 64 + 64) * sizeof(float), stream);  // gsum + gcnt contiguous
    {
        long long tot = (long long)N_NODES * 64;
        pool_kernel<<<(int)((tot + B - 1) / B), B, 0, stream>>>(h3, batch, gsum, gcnt);
    }
    head_kernel<<<1, 256, 0, stream>>>(gsum, gcnt, Wl1, bl1, Wl2, bl2, Wl3, bl3, (float*)d_out);
}